// YoloLoss_35562329211712
// MI455X (gfx1250) — compile-verified
//
#include <hip/hip_runtime.h>
#include <math.h>

// ---------------- types for CDNA5 builtins ----------------
typedef float v2f  __attribute__((ext_vector_type(2)));
typedef float v8f  __attribute__((ext_vector_type(8)));
typedef unsigned int u32x4 __attribute__((ext_vector_type(4)));
typedef int   i32x4 __attribute__((ext_vector_type(4)));
typedef int   i32x8 __attribute__((ext_vector_type(8)));

// ---------------- problem constants (from reference) ----------------
#define Bn   16
#define Hn   52
#define Wn   52
#define An   5
#define Mn   40            // boxes per batch
#define HWA  (Hn*Wn*An)    // 13520 cells per batch image
#define BLK  256
#define GX   ((HWA + BLK - 1)/BLK)   // 53
#define NPART (GX*Bn)                // 848 block partials

__device__ __forceinline__ float sigmoidf_(float x) {
  return 1.0f / (1.0f + __expf(-x));
}

// Wave32 sum-reduction using V_WMMA_F32_16X16X4_F32.
// A(16x4): VGPR a.x -> A[m,0] (lanes 0..15) / A[m,2] (lanes 16..31); a.y -> K=1/3 (zeroed).
// B(4x16) = ones  =>  D[m,n] = p[m] + p[m+16]  (replicated over n).
// Lane<16 holds D rows 0..7 in c[0..7]; lane>=16 holds rows 8..15.
// Sum of 8 VGPRs = half-wave sum; ds_swizzle SWAPX16 (xor 0x10) combines halves.
__device__ __forceinline__ float wave_reduce_sum(float v) {
  v2f a; a.x = v;    a.y = 0.0f;
  v2f b; b.x = 1.0f; b.y = 1.0f;
  v8f c = {0.f,0.f,0.f,0.f,0.f,0.f,0.f,0.f};
  c = __builtin_amdgcn_wmma_f32_16x16x4_f32(false, a, false, b, (short)0, c, false, false);
  float t = ((c[0] + c[1]) + (c[2] + c[3])) + ((c[4] + c[5]) + (c[6] + c[7]));
  // group-of-32 swizzle: xor_mask=0x10, and_mask=0x1f  -> 0x401F (SWAPX16)
  float o = __int_as_float(__builtin_amdgcn_ds_swizzle(__float_as_int(t), 0x401F));
  return t + o;
}

__global__ __launch_bounds__(BLK)
void yolo_main(const float* __restrict__ outputs,
               const float* __restrict__ samp,
               const float* __restrict__ ytrue,
               const float* __restrict__ anchors,
               const float* __restrict__ thr,
               const float* __restrict__ scalez,
               float* __restrict__ partials) {
  __shared__ float raw[Mn * 5];      // TDM destination: raw samp_bndbxs[b]
  __shared__ float box[Mn * 5];      // {tminx,tminy,tmaxx,tmaxy,tarea}
  __shared__ float wsum[BLK / 32];

  const int b = blockIdx.y;

  // ---- Stage samp_bndbxs[b] (200 f32 = 800B) into LDS via the Tensor Data Mover.
  // 1-D descriptor: tensor_dim0 = tile_dim0 = 200 elements of data_size 4B.
  if (threadIdx.x < 32) {   // wave 0 only issues the TDM op (EXEC ignored by TDM)
    unsigned long long ga =
        (unsigned long long)(size_t)(samp + (size_t)b * (Mn * 5));
    unsigned lds_off = (unsigned)(unsigned long long)(size_t)(&raw[0]); // low 32b = LDS offset
    u32x4 g0;
    g0[0] = 1u;                                   // count=1, user descriptor
    g0[1] = lds_off;                              // lds_addr
    g0[2] = (unsigned)(ga & 0xFFFFFFFFu);         // global_addr[31:0]
    g0[3] = (unsigned)((ga >> 32) & 0x01FFFFFFu)  // global_addr[56:32]
          | (2u << 30);                           // type=2 ("image")
    i32x8 g1;
    g1[0] = (int)(2u << 16);                      // data_size=2 (4 bytes)
    g1[1] = (int)(((unsigned)(Mn * 5)) << 16);    // tensor_dim0[15:0] @ bits[63:48]
    g1[2] = 0;                                    // tensor_dim0 hi / tensor_dim1
    g1[3] = (int)(((unsigned)(Mn * 5)) << 16);    // tile_dim0 @ bits[127:112]
    g1[4] = 0;                                    // tile_dim1/2 unused
    g1[5] = (int)(Mn * 5);                        // tensor_dim0_stride[31:0]
    g1[6] = 0;
    g1[7] = 0;
    i32x4 z4 = {0, 0, 0, 0};
    i32x8 z8 = {0, 0, 0, 0, 0, 0, 0, 0};
    __builtin_amdgcn_tensor_load_to_lds(g0, g1, z4, z4, z8, 0);
    __builtin_amdgcn_s_wait_tensorcnt(0);
  }
  __syncthreads();

  // ---- Build per-box records once per block.
  // Masked-out boxes get (+inf,+inf,-inf,-inf,0): the IoU math then yields exactly 0.
  if (threadIdx.x < Mn) {
    const int m = threadIdx.x;
    float e1 = raw[m * 5 + 1], e2 = raw[m * 5 + 2];
    float e3 = raw[m * 5 + 3], e4 = raw[m * 5 + 4];
    e2 = (fabsf(e2) == __builtin_inff()) ? 0.0f : e2;   // jnp.where(isinf(col2),0,col2)
    const bool bm = e2 > 1e-4f;
    const float inf = __builtin_inff();
    box[m * 5 + 0] = bm ? e1 - 0.5f * e3 : inf;
    box[m * 5 + 1] = bm ? e2 - 0.5f * e4 : inf;
    box[m * 5 + 2] = bm ? e1 + 0.5f * e3 : -inf;
    box[m * 5 + 3] = bm ? e2 + 0.5f * e4 : -inf;
    box[m * 5 + 4] = bm ? e3 * e4 : 0.0f;
  }
  __syncthreads();

  const int hwa = blockIdx.x * BLK + threadIdx.x;
  float partial = 0.0f;

  if (hwa < HWA) {
    const int a = hwa % An;
    const int w = (hwa / An) % Wn;
    const int h = hwa / (An * Wn);

    const float* o = outputs + (((size_t)b * Hn + h) * Wn + w) * (An * 6) + a * 6;
    const float f0 = o[0], f1 = o[1], f2 = o[2], f3 = o[3], f4 = o[4], f5 = o[5];
    const float anx = anchors[a * 2 + 0], any_ = anchors[a * 2 + 1];

    const float sx = sigmoidf_(f0), sy = sigmoidf_(f1);
    const float cf = sigmoidf_(f4), cl = sigmoidf_(f5);

    const float invW = 1.0f / (float)Wn, invH = 1.0f / (float)Hn;
    const float px  = (sx + (float)w) * invW;
    const float py  = (sy + (float)h) * invH;
    const float pwx = __expf(f2) * anx * invW;
    const float pwy = __expf(f3) * any_ * invH;
    const float pmx = px - 0.5f * pwx, pMx = px + 0.5f * pwx;
    const float pmy = py - 0.5f * pwy, pMy = py + 0.5f * pwy;
    const float pa  = pwx * pwy;

    // ---- best IoU over 40 boxes, division-free (track numerator/denominator).
    float bia = 0.0f, bden = 1.0f;     // represents best IoU = 0
#pragma unroll 5
    for (int m = 0; m < Mn; ++m) {
      const float bx0 = box[m * 5 + 0], by0 = box[m * 5 + 1];
      const float bx1 = box[m * 5 + 2], by1 = box[m * 5 + 3];
      const float ta  = box[m * 5 + 4];
      float ix = fminf(pMx, bx1) - fmaxf(pmx, bx0);
      float iy = fminf(pMy, by1) - fmaxf(pmy, by0);
      ix = fmaxf(ix, 0.0f);
      iy = fmaxf(iy, 0.0f);
      const float ia  = ix * iy;
      const float den = pa + ta - ia + 1e-5f;
      const bool upd = ia * bden > bia * den;  // ia/den > bia/bden, both dens > 0
      bia  = upd ? ia  : bia;
      bden = upd ? den : bden;
    }
    const float noones = (bia < thr[0] * bden) ? 1.0f : 0.0f;

    // ---- y_true loss terms (mirrors reference unit conventions exactly).
    const float* yt = ytrue + ((((size_t)b * Hn + h) * Wn + w) * An + a) * 5;
    const float yx = yt[0], yy = yt[1], yw = yt[2], yh = yt[3], ones = yt[4];

    const float tbx = (yx + (float)w) * invW;
    const float tby = (yy + (float)h) * invH;
    const float twx = __logf(yw * (float)Wn / anx + 1e-6f);
    const float twy = __logf(yh * (float)Hn / any_ + 1e-6f);
    const float tmx = yx - 0.5f * yw, tMx = yx + 0.5f * yw;
    const float tmy = yy - 0.5f * yh, tMy = yy + 0.5f * yh;

    const float iwx = fmaxf(fminf(pMx, tMx) - fmaxf(pmx, tmx), 0.0f);
    const float iwy = fmaxf(fminf(pMy, tMy) - fmaxf(pmy, tmy), 0.0f);
    const float ia1 = iwx * iwy;
    const float tareas = yw * yh;
    const float iou_s  = ia1 / (pa + tareas - ia1 + 1e-5f);

    const float s_obj = scalez[0], s_nob = scalez[1];
    const float s_cls = scalez[2], s_crd = scalez[3];

    const float dcf = iou_s - cf;
    const float dcl = 1.0f - cl;
    const float dxy = (tbx - sx) * (tbx - sx) + (tby - sy) * (tby - sy);
    const float dwh = (twx - f2) * (twx - f2) + (twy - f3) * (twy - f3);

    partial = dcf * dcf * ones * s_obj
            + cf  * cf  * noones * s_nob
            + dcl * dcl * ones * s_cls
            + dxy * ones * s_crd
            + dwh * ones * ones;
  }

  // ---- block reduction: WMMA wave-sum, then cross-wave via LDS (no atomics).
  const float wred = wave_reduce_sum(partial);   // EXEC all-ones here: no divergence
  const int lane = threadIdx.x & 31, wid = threadIdx.x >> 5;
  if (lane == 0) wsum[wid] = wred;
  __syncthreads();
  if (threadIdx.x == 0) {
    float r = 0.0f;
#pragma unroll
    for (int i = 0; i < BLK / 32; ++i) r += wsum[i];
    partials[blockIdx.y * gridDim.x + blockIdx.x] = r;
  }
}

__global__ __launch_bounds__(BLK)
void yolo_reduce(const float* __restrict__ partials, float* __restrict__ out, int n) {
  __shared__ float wsum[BLK / 32];
  float v = 0.0f;
  for (int i = threadIdx.x; i < n; i += BLK) v += partials[i];
  const float wred = wave_reduce_sum(v);
  const int lane = threadIdx.x & 31, wid = threadIdx.x >> 5;
  if (lane == 0) wsum[wid] = wred;
  __syncthreads();
  if (threadIdx.x == 0) {
    float r = 0.0f;
#pragma unroll
    for (int i = 0; i < BLK / 32; ++i) r += wsum[i];
    out[0] = r;   // plain store: poisoned d_out needs no init pass
  }
}

extern "C" void kernel_launch(void* const* d_in, const int* in_sizes, int n_in,
                              void* d_out, int out_size, void* d_ws, size_t ws_size,
                              hipStream_t stream) {
  (void)in_sizes; (void)n_in; (void)out_size; (void)ws_size;
  const float* outputs = (const float*)d_in[0];   // (16,52,52,30)
  const float* samp    = (const float*)d_in[1];   // (16,40,5)
  const float* ytrue   = (const float*)d_in[2];   // (16,52,52,5,5)
  const float* anchors = (const float*)d_in[3];   // (5,2)
  const float* thr     = (const float*)d_in[4];   // (1,)
  const float* scalez  = (const float*)d_in[5];   // (4,)
  // d_in[6] = cell_grid: recomputed analytically as (w,h) in-kernel.

  float* partials = (float*)d_ws;                 // 848 f32 block partials

  dim3 grid(GX, Bn);
  yolo_main<<<grid, BLK, 0, stream>>>(outputs, samp, ytrue, anchors, thr, scalez, partials);
  yolo_reduce<<<1, BLK, 0, stream>>>(partials, (float*)d_out, NPART);
}